// HLfilter_construct_73315091742991
// MI455X (gfx1250) — compile-verified
//
#include <hip/hip_runtime.h>
#include <hip/hip_bf16.h>

#define NDIM 8192

typedef float v2f __attribute__((ext_vector_type(2)));
typedef float v4f __attribute__((ext_vector_type(4)));
typedef float v8f __attribute__((ext_vector_type(8)));

// ---------------------------------------------------------------------------
// Pass 1: row sums of A via V_WMMA_F32_16X16X4_F32  (D = A_tile(16x4) x ones(4x16) + C).
// One wave handles 16 rows x 1024 K-columns (K chunk); 8 chunks per row-group
// -> 4096 waves. A-fragment layout (ISA 7.12.2, 32-bit A 16x4):
//   lane m (0..15):  v0,v1 = A[m][k+0], A[m][k+1]
//   lane m+16:       v0,v1 = A[m][k+2], A[m][k+3]
// so each lane does sequential float2 loads from its own row (cache-friendly
// streaming). ones-B is layout independent. D[m][n] == rowsum(m) for all n;
// lane 0 holds M=0..7 in c[0..7], lane 16 holds M=8..15.
// ---------------------------------------------------------------------------
__global__ void __launch_bounds__(256) rowsum_wmma_kernel(const float* __restrict__ A,
                                                          float* __restrict__ partial) {
    const int lane  = threadIdx.x & 31;
    const int wid   = (blockIdx.x << 3) + (threadIdx.x >> 5);   // 8 waves / block
    const int g     = wid >> 3;                                 // row group [0,512)
    const int chunk = wid & 7;                                  // K chunk  [0,8)
    const int row   = (g << 4) + (lane & 15);
    const int col0  = (chunk << 10) + ((lane >> 4) << 1);
    const float* p  = A + (size_t)row * NDIM + col0;

    v8f c = {0.f, 0.f, 0.f, 0.f, 0.f, 0.f, 0.f, 0.f};
    const v2f b = {1.f, 1.f};   // all-ones B fragment (4x16)

    for (int k0 = 0; k0 < 1024; k0 += 64) {
        // stream-prefetch 2KB ahead in this lane's row (global_prefetch_b8)
        __builtin_prefetch(p + k0 + 512, 0, 3);
#pragma unroll
        for (int k = k0; k < k0 + 64; k += 4) {
            v2f a = *(const v2f*)(p + k);
            c = __builtin_amdgcn_wmma_f32_16x16x4_f32(
                    /*neg_a=*/false, a, /*neg_b=*/false, b,
                    /*c_mod=*/(short)0, c, /*reuse_a=*/false, /*reuse_b=*/false);
        }
    }

    if ((lane & 15) == 0) {                 // lanes 0 and 16 hold the sums
        const int mbase = (lane >> 4) << 3; // lane0 -> rows 0..7, lane16 -> 8..15
#pragma unroll
        for (int r = 0; r < 8; ++r)
            partial[(size_t)((g << 4) + mbase + r) * 8 + chunk] = c[r];
    }
}

// ---------------------------------------------------------------------------
// Pass 1b: fold the 8 deterministic K-chunk partials, d = 1/sqrt(sum), inf->0.
// ---------------------------------------------------------------------------
__global__ void __launch_bounds__(256) finalize_d_kernel(const float* __restrict__ partial,
                                                         float* __restrict__ dvec) {
    const int row = blockIdx.x * blockDim.x + threadIdx.x;   // 8192 threads
    const float* p = partial + (size_t)row * 8;
    float s = 0.f;
#pragma unroll
    for (int r = 0; r < 8; ++r) s += p[r];
    float dv = 1.0f / __builtin_sqrtf(s);
    if (__builtin_isinf(dv)) dv = 0.0f;
    dvec[row] = dv;
}

// ---------------------------------------------------------------------------
// Pass 2: elementwise, float4 per thread.
//   L = d[i]*A*d[j];  H = (i==j ? 2 : 0) - L;  copy A.
// A read and all three output streams are touched exactly once -> NT hints,
// keeping the 32KB d vector resident in cache instead.
// ---------------------------------------------------------------------------
__global__ void __launch_bounds__(256) dad_filters_kernel(const float* __restrict__ A,
                                                          const float* __restrict__ dvec,
                                                          float* __restrict__ out) {
    const long long NN = (long long)NDIM * NDIM;
    const long long e  = ((long long)blockIdx.x * blockDim.x + threadIdx.x) * 4LL;
    const int i = (int)(e >> 13);        // row (NDIM == 2^13)
    const int j = (int)(e & (NDIM - 1)); // col of first element

    v4f a  = __builtin_nontemporal_load((const v4f*)(A + e));
    float di = dvec[i];                  // temporal: reused by whole row
    v4f dr = *(const v4f*)(dvec + j);    // temporal: reused by every row

    v4f l = (di * dr) * a;
    v4f h = -l;
    if (i >= j && i < j + 4) h[i - j] += 2.0f;   // diagonal of 2I - DAD

    __builtin_nontemporal_store(l, (v4f*)(out + e));            // Lfilter
    __builtin_nontemporal_store(h, (v4f*)(out + NN + e));       // Hfilter
    __builtin_nontemporal_store(a, (v4f*)(out + 2 * NN + e));   // Graph_adj copy
}

// ---------------------------------------------------------------------------
extern "C" void kernel_launch(void* const* d_in, const int* in_sizes, int n_in,
                              void* d_out, int out_size, void* d_ws, size_t ws_size,
                              hipStream_t stream) {
    const float* A = (const float*)d_in[0];
    float* out     = (float*)d_out;
    float* ws      = (float*)d_ws;
    float* dvec    = ws;            // 8192 floats
    float* partial = ws + NDIM;     // 8192*8 floats

    // Pass 1: 512 row-groups x 8 K-chunks = 4096 waves = 512 blocks x 8 waves.
    rowsum_wmma_kernel<<<512, 256, 0, stream>>>(A, partial);

    // Pass 1b: 8192 rows.
    finalize_d_kernel<<<NDIM / 256, 256, 0, stream>>>(partial, dvec);

    // Pass 2: N*N/4 float4 threads = 16M threads = 65536 blocks x 256.
    const long long nThreads = (long long)NDIM * NDIM / 4;
    dad_filters_kernel<<<(unsigned)(nThreads / 256), 256, 0, stream>>>(A, dvec, out);
}